// MoSRAHRouter_23802708754603
// MI455X (gfx1250) — compile-verified
//
#include <hip/hip_runtime.h>
#include <math.h>

// Problem constants (from reference): B=4, N=8192, H=2048, L=64, K=8
#define B_    4
#define N_    8192
#define H_    2048
#define L_    64
#define K_    8
#define NTOK  (B_ * N_)        // 32768 tokens
#define KCHUNK 128             // K-slab of W staged in LDS (128*64*4 = 32 KB)

typedef __attribute__((ext_vector_type(2))) float v2f;
typedef __attribute__((ext_vector_type(8))) float v8f;

// ---------------------------------------------------------------------------
// Kernel 1: logits[M=32768, L=64] = x[M, H=2048] @ W[H, L]
// V_WMMA_F32_16X16X4_F32. 8 waves/block, one 16-row tile per wave.
//
// A 16x4 fp32 fragment (ISA 7.12.2): lanes 0-15 rows M=0..15 carry (K0,K1),
// lanes 16-31 carry (K2,K3) -> contiguous per-lane float2 at (row, k0+2*half).
// B 4x16 fragment: VGPR0 = row K0 (lanes 0-15) / K2 (lanes 16-31),
// VGPR1 = K1 / K3, N = lane&15 -> per-lane pair (W[k][n], W[k+1][n]).
//
// W slab is staged into LDS *k-pair interleaved*: lw[kpair][n] = float2
// (W[2*kpair][n], W[2*kpair+1][n]). Each B fragment is then a single
// 8B-aligned ds_load_b64 straight into an even VGPR pair - no repack movs.
// C/D 16x16: VGPR r holds (M = r + 8*half, N = lane&15).
// ---------------------------------------------------------------------------
__global__ __launch_bounds__(256)
void logits_wmma_kernel(const float* __restrict__ x,
                        const float* __restrict__ W,
                        float* __restrict__ logits) {
  __shared__ v2f lw[(KCHUNK / 2) * L_];            // 64 k-pairs x 64 n = 32 KB

  const int lane = threadIdx.x & 31;
  const int wave = threadIdx.x >> 5;
  const int m0   = (blockIdx.x * 8 + wave) * 16;   // 16-token tile base
  const int half = lane >> 4;                      // 0: lanes 0-15, 1: 16-31
  const int l16  = lane & 15;

  v8f acc[4] = {v8f{}, v8f{}, v8f{}, v8f{}};       // 4 N-tiles of 16 experts

  const float* xrow = x + (size_t)(m0 + l16) * H_ + half * 2;

  for (int kc = 0; kc < H_; kc += KCHUNK) {
    __syncthreads();                               // previous slab consumed
    {
      // Cooperative slab load with k-pair interleave.
      // 64 kpairs x 16 n-groups of 4 = 1024 units over 256 threads.
      #pragma unroll
      for (int i = 0; i < 4; ++i) {
        const int g     = threadIdx.x + i * 256;
        const int kpair = g >> 4;
        const int ng    = (g & 15) * 4;
        const float4 r0 = *(const float4*)(W + (size_t)(kc + 2 * kpair) * L_ + ng);
        const float4 r1 = *(const float4*)(W + (size_t)(kc + 2 * kpair + 1) * L_ + ng);
        float4* dst = (float4*)(lw + kpair * L_ + ng);
        dst[0] = make_float4(r0.x, r1.x, r0.y, r1.y);
        dst[1] = make_float4(r0.z, r1.z, r0.w, r1.w);
      }
    }
    __syncthreads();

    // Prefetch next slab of this wave's A stream (WGP-scope temporal).
    if (kc + KCHUNK < H_) __builtin_prefetch(xrow + kc + KCHUNK, 0, 3);

    #pragma unroll 4
    for (int kk = 0; kk < KCHUNK; kk += 4) {
      const v2f a = *(const v2f*)(xrow + kc + kk);         // (k0+2h, k0+1+2h)
      const v2f* wp = lw + ((kk >> 1) + half) * L_ + l16;  // pair row for this half
      #pragma unroll
      for (int nt = 0; nt < 4; ++nt) {
        const v2f b = wp[nt * 16];                         // single ds_load_b64
        acc[nt] = __builtin_amdgcn_wmma_f32_16x16x4_f32(
            /*neg_a=*/false, a, /*neg_b=*/false, b,
            /*c_mod=*/(short)0, acc[nt],
            /*reuse_a=*/false, /*reuse_b=*/false);
      }
    }
  }

  // Store: VGPR r holds (M = m0 + r + 8*half, N = l16) per N-tile.
  #pragma unroll
  for (int r = 0; r < 8; ++r) {
    float* orow = logits + (size_t)(m0 + r + half * 8) * L_ + l16;
    orow[0]  = acc[0][r];
    orow[16] = acc[1][r];
    orow[32] = acc[2][r];
    orow[48] = acc[3][r];
  }
}

// ---------------------------------------------------------------------------
// Kernel 2: per-token routing. One wave32 per token; 2 experts per lane.
// top-k(softmax(logits+bias)) == top-k(logits+bias) (softmax is monotone),
// and gathered/sum(gathered) cancels the softmax denominator, so only the
// selected exp(logit - rowmax) values are needed.
// ---------------------------------------------------------------------------
__global__ __launch_bounds__(256)
void route_kernel(const float* __restrict__ logits,
                  const float* __restrict__ bias,
                  float* __restrict__ out_idx,
                  float* __restrict__ out_prob,
                  float* __restrict__ counts) {
  const int lane = threadIdx.x & 31;
  const int wave = threadIdx.x >> 5;
  const size_t t = (size_t)blockIdx.x * 8 + wave;  // token id < 32768

  const float* lrow = logits + t * L_;
  const float l0 = lrow[lane];
  const float l1 = lrow[lane + 32];
  float bb0 = l0 + bias[lane];
  float bb1 = l1 + bias[lane + 32];

  // Row max of unbiased logits (numerical stability).
  float m = fmaxf(l0, l1);
  #pragma unroll
  for (int off = 16; off >= 1; off >>= 1) m = fmaxf(m, __shfl_xor(m, off));

  const float e0 = __expf(l0 - m);
  const float e1 = __expf(l1 - m);

  // Iterative wave-argmax top-8 on biased logits; lower index wins ties
  // (matches jax.lax.top_k).
  int sel[K_];
  #pragma unroll
  for (int k = 0; k < K_; ++k) {
    float bv; int bi;
    if (bb0 >= bb1) { bv = bb0; bi = lane; } else { bv = bb1; bi = lane + 32; }
    #pragma unroll
    for (int off = 16; off >= 1; off >>= 1) {
      float ov = __shfl_xor(bv, off);
      int   oi = __shfl_xor(bi, off);
      if (ov > bv || (ov == bv && oi < bi)) { bv = ov; bi = oi; }
    }
    sel[k] = bi;                                   // uniform across the wave
    if (bi == lane)      bb0 = -__builtin_inff();
    if (bi == lane + 32) bb1 = -__builtin_inff();
  }

  // Gather unbiased exp values at the selected experts; renormalize.
  float g[K_], gsum = 0.0f;
  #pragma unroll
  for (int k = 0; k < K_; ++k) {
    const int j = sel[k];
    const float ge = (j < 32) ? __shfl(e0, j) : __shfl(e1, j - 32);
    g[k] = ge;
    gsum += ge;
  }
  const float inv = 1.0f / gsum;

  if (lane < K_) {
    out_idx [t * K_ + lane] = (float)sel[lane];
    out_prob[t * K_ + lane] = g[lane] * inv;
    atomicAdd(&counts[sel[lane]], 1.0f);           // global_atomic_add_f32
  }
}

// ---------------------------------------------------------------------------
// Kernel 3: lb_loss = sum_l |counts[l]/(B*N*K) - 1/L|. One wave.
// ---------------------------------------------------------------------------
__global__ void loss_kernel(const float* __restrict__ counts,
                            float* __restrict__ out_loss) {
  const int lane = threadIdx.x & 31;
  const float inv = 1.0f / (float)(NTOK * K_);     // 1/262144
  float v = fabsf(counts[lane] * inv - 1.0f / (float)L_) +
            fabsf(counts[lane + 32] * inv - 1.0f / (float)L_);
  #pragma unroll
  for (int off = 16; off >= 1; off >>= 1) v += __shfl_xor(v, off);
  if (lane == 0) out_loss[0] = v;
}

__global__ void zero_counts_kernel(float* __restrict__ counts) {
  counts[threadIdx.x] = 0.0f;                      // 64 threads
}

// ---------------------------------------------------------------------------
// Launch: d_in = {x (B,N,H) f32, W (H,L) f32, expert_bias (L,) f32}
// d_out = [selected_heads (B,N,K)] ++ [routing_probs (B,N,K)] ++ [lb_loss]
// d_ws  = [logits: 32768*64 f32 = 8 MB][counts: 64 f32]
// ---------------------------------------------------------------------------
extern "C" void kernel_launch(void* const* d_in, const int* in_sizes, int n_in,
                              void* d_out, int out_size, void* d_ws, size_t ws_size,
                              hipStream_t stream) {
  const float* x    = (const float*)d_in[0];
  const float* W    = (const float*)d_in[1];
  const float* bias = (const float*)d_in[2];

  float* out      = (float*)d_out;
  float* out_idx  = out;
  float* out_prob = out + (size_t)NTOK * K_;
  float* out_loss = out + (size_t)2 * NTOK * K_;

  float* logits = (float*)d_ws;
  float* counts = logits + (size_t)NTOK * L_;

  zero_counts_kernel<<<1, L_, 0, stream>>>(counts);
  logits_wmma_kernel<<<NTOK / 128, 256, 0, stream>>>(x, W, logits);
  route_kernel<<<NTOK / 8, 256, 0, stream>>>(logits, bias, out_idx, out_prob, counts);
  loss_kernel<<<1, 32, 0, stream>>>(counts, out_loss);
}